// RepSet_15960098472409
// MI455X (gfx1250) — compile-verified
//
#include <hip/hip_runtime.h>
#include <hip/hip_bf16.h>

// RepSet: out[b,o] = sum_s max_h relu( Xs[b,s,:] . W[o,h,:] )
// B=64, S=128, E=64, O=256, H=64  (S > H => max over h, sum over s)
//
// Block = (batch b, output pair {o0,o1}), 256 threads = 8 waves.
// Waves 0-3 -> o0, waves 4-7 -> o1; each wave owns 32 s-rows (two 16-row
// M-tiles) so every B fragment loaded from LDS feeds TWO v_wmma_f32_16x16x4_f32.

typedef float v2f __attribute__((ext_vector_type(2)));
typedef float v8f __attribute__((ext_vector_type(8)));
typedef int   v4i __attribute__((ext_vector_type(4)));

#define BATCH     64
#define SET_SIZE  128
#define EDIM      64
#define ODIM      256
#define HDIM      64

// W-tile LDS row stride: 64 + 4 pad floats => bank stride 4 mod 64, so the
// per-lane B-fragment b64 loads (banks {4*l16 + 2*half, +1}) are conflict-free.
#define WSTRIDE     68
#define TILE_FLOATS (HDIM * WSTRIDE)

#if __has_builtin(__builtin_amdgcn_global_load_async_to_lds_b128)
#define HAVE_ASYNC_LDS 1
typedef __attribute__((address_space(1))) v4i g_v4i;
typedef __attribute__((address_space(3))) v4i l_v4i;
__device__ __forceinline__ void async_copy_b128(const float* gsrc, const float* ldst) {
    __builtin_amdgcn_global_load_async_to_lds_b128(
        (g_v4i*)(unsigned long long)(const void*)gsrc,
        (l_v4i*)(unsigned)(unsigned long long)(const void*)ldst,
        /*offset=*/0, /*cpol=*/0);
}
__device__ __forceinline__ void async_wait0() {
#if __has_builtin(__builtin_amdgcn_s_wait_asynccnt)
    __builtin_amdgcn_s_wait_asynccnt(0);
#else
    asm volatile("s_wait_asynccnt 0" ::: "memory");
#endif
}
#else
#define HAVE_ASYNC_LDS 0
#endif

__global__ __launch_bounds__(256) void repset_wmma_f32(
    const float* __restrict__ Xs,   // (B, S, E)
    const float* __restrict__ W,    // (O, H, E)
    float* __restrict__ out)        // (B, O)
{
    __shared__ float w_lds[2 * TILE_FLOATS];
    __shared__ float red[16];

    const int b    = blockIdx.x;      // 0..63
    const int op   = blockIdx.y;      // 0..127 : output pair
    const int tid  = threadIdx.x;
    const int wave = tid >> 5;        // 0..7
    const int og   = wave >> 2;       // 0/1 : which o of the pair
    const int wq   = wave & 3;        // 0..3 : s-quarter (32 rows)
    const int lane = tid & 31;
    const int half = lane >> 4;       // K/M split per WMMA layout
    const int l16  = lane & 15;

    // ---- stage W[o0] and W[o1] (2 x 64x64 f32) into padded LDS ----
#if HAVE_ASYNC_LDS
    // 2048 b128 chunks (2 tiles x 64 rows x 16 chunks), 8 per thread,
    // consecutive tids -> consecutive 16B chunks (fully coalesced).
    #pragma unroll
    for (int i = 0; i < 8; ++i) {
        const int c   = tid + 256 * i;
        const int t   = c >> 10;        // tile 0/1
        const int row = (c >> 4) & 63;  // 0..63
        const int seg = c & 15;         // 0..15
        const float* gsrc = W + ((size_t)(2 * op + t)) * (HDIM * EDIM)
                              + row * EDIM + seg * 4;
        const float* ldst = &w_lds[t * TILE_FLOATS + row * WSTRIDE + seg * 4];
        async_copy_b128(gsrc, ldst);
    }
#else
    {
        const int row  = tid >> 4;          // 0..15
        const int col4 = (tid & 15) << 2;   // 0,4,...,60
        #pragma unroll
        for (int t = 0; t < 2; ++t) {
            const float* wsrc = W + ((size_t)(2 * op + t)) * (HDIM * EDIM);
            #pragma unroll
            for (int i = 0; i < 4; ++i) {
                float4 v = *(const float4*)(wsrc + (row + 16 * i) * EDIM + col4);
                float* dst = &w_lds[t * TILE_FLOATS + (row + 16 * i) * WSTRIDE + col4];
                dst[0] = v.x; dst[1] = v.y; dst[2] = v.z; dst[3] = v.w;
            }
        }
    }
#endif

    // ---- preload A fragments for this wave's TWO 16-row M-tiles ----
    // A 16x4 f32 layout: lanes 0-15 hold {K=0,K=1}, lanes 16-31 hold {K=2,K=3}:
    // K-step k reads Xs[b][s][4k + 2*half .. +1] (contiguous v2f).
    const float* x0 = Xs + ((size_t)b * SET_SIZE + wq * 32 + l16) * EDIM + 2 * half;
    v2f a0[16], a1[16];
    #pragma unroll
    for (int k = 0; k < 16; ++k) {
        a0[k] = *(const v2f*)(x0 + 4 * k);
        a1[k] = *(const v2f*)(x0 + 16 * EDIM + 4 * k);
    }

#if HAVE_ASYNC_LDS
    async_wait0();            // this wave's async chunks are in LDS
#endif
    __syncthreads();          // all waves' chunks visible

    // ---- K-loop: 16 steps x (4 h-tiles x 2 M-tiles) = 128 WMMAs per wave ----
    v8f a00 = {}, a01 = {}, a02 = {}, a03 = {};   // M-tile 0, h-tiles 0..3
    v8f a10 = {}, a11 = {}, a12 = {}, a13 = {};   // M-tile 1, h-tiles 0..3
    const float* bbase = &w_lds[og * TILE_FLOATS + l16 * WSTRIDE + 2 * half];
    #pragma unroll
    for (int k = 0; k < 16; ++k) {
        v2f b0 = *(const v2f*)(bbase + 4 * k);
        v2f b1 = *(const v2f*)(bbase + 16 * WSTRIDE + 4 * k);
        v2f b2 = *(const v2f*)(bbase + 32 * WSTRIDE + 4 * k);
        v2f b3 = *(const v2f*)(bbase + 48 * WSTRIDE + 4 * k);
        a00 = __builtin_amdgcn_wmma_f32_16x16x4_f32(false, a0[k], false, b0, (short)0, a00, false, false);
        a10 = __builtin_amdgcn_wmma_f32_16x16x4_f32(false, a1[k], false, b0, (short)0, a10, false, false);
        a01 = __builtin_amdgcn_wmma_f32_16x16x4_f32(false, a0[k], false, b1, (short)0, a01, false, false);
        a11 = __builtin_amdgcn_wmma_f32_16x16x4_f32(false, a1[k], false, b1, (short)0, a11, false, false);
        a02 = __builtin_amdgcn_wmma_f32_16x16x4_f32(false, a0[k], false, b2, (short)0, a02, false, false);
        a12 = __builtin_amdgcn_wmma_f32_16x16x4_f32(false, a1[k], false, b2, (short)0, a12, false, false);
        a03 = __builtin_amdgcn_wmma_f32_16x16x4_f32(false, a0[k], false, b3, (short)0, a03, false, false);
        a13 = __builtin_amdgcn_wmma_f32_16x16x4_f32(false, a1[k], false, b3, (short)0, a13, false, false);
    }

    // ---- epilogue: relu + max over 64 h, sum over this wave's 32 s-rows ----
    // C layout: VGPR v, lane = (row M = v + 8*half, col N = l16).
    // elementwise max over h-tiles => max over {N, N+16, N+32, N+48};
    // 16-lane xor-max (masks 1,2,4,8 stay in each half) => max over all 64 h.
    float rowsum = 0.0f;
    #pragma unroll
    for (int v = 0; v < 8; ++v) {
        float m0 = fmaxf(fmaxf(a00[v], a01[v]), fmaxf(a02[v], a03[v]));
        float m1 = fmaxf(fmaxf(a10[v], a11[v]), fmaxf(a12[v], a13[v]));
        m0 = fmaxf(m0, 0.0f);           // relu commutes with max
        m1 = fmaxf(m1, 0.0f);
        #pragma unroll
        for (int s = 1; s < 16; s <<= 1) {
            m0 = fmaxf(m0, __shfl_xor(m0, s, 32));
            m1 = fmaxf(m1, __shfl_xor(m1, s, 32));
        }
        rowsum += m0 + m1;              // rowmax for rows v(+8*half) of both M-tiles
    }

    if (l16 == 0) red[wave * 2 + half] = rowsum;   // waves 0-3 -> [0..7], 4-7 -> [8..15]
    __syncthreads();

    if (tid == 0) {
        float t = 0.0f;
        #pragma unroll
        for (int i = 0; i < 8; ++i) t += red[i];
        out[(size_t)b * ODIM + 2 * op + 0] = t;
    }
    if (tid == 128) {
        float t = 0.0f;
        #pragma unroll
        for (int i = 8; i < 16; ++i) t += red[i];
        out[(size_t)b * ODIM + 2 * op + 1] = t;
    }
}

extern "C" void kernel_launch(void* const* d_in, const int* in_sizes, int n_in,
                              void* d_out, int out_size, void* d_ws, size_t ws_size,
                              hipStream_t stream) {
    (void)in_sizes; (void)n_in; (void)out_size; (void)d_ws; (void)ws_size;
    const float* Xs = (const float*)d_in[0];
    const float* W  = (const float*)d_in[1];
    float* out      = (float*)d_out;

    dim3 grid(BATCH, ODIM / 2);   // one workgroup per (b, {o0,o1})
    repset_wmma_f32<<<grid, 256, 0, stream>>>(Xs, W, out);
}